// SplineRegionalBinaryEncoding2_75969381532164
// MI455X (gfx1250) — compile-verified
//
#include <hip/hip_runtime.h>
#include <hip/hip_bf16.h>

// Problem constants (from the reference):
//   WINDOW = (-100000, 100000)
//   BINWIDTHS  = {100, 200, 500, 1000, 2000, 5000}
//   NBINS      = {2001, 1001, 401, 201, 101, 41}   (sum = 3746 = N_WEIGHTS)
//   BINCUMSTART= {0, 2001, 3002, 3403, 3604, 3705}
//   F=100000, C=2, R=500, D=100
#define F_COUNT   100000
#define D_DIM     100
#define NW        3746
#define SEGS      25              // D_DIM / 4 float4 segments
#define TOTAL_THREADS (F_COUNT * SEGS)

__global__ __launch_bounds__(256) void spline_gather_kernel(
    const int*   __restrict__ coords,     // [F,2] int32
    const int*   __restrict__ region_ids, // [F]   int32
    const float* __restrict__ W,          // [R, NW, D] f32
    const float* __restrict__ bias,       // [R, D]     f32
    float*       __restrict__ out)        // [F, D]     f32
{
    int tid = blockIdx.x * blockDim.x + threadIdx.x;
    if (tid >= TOTAL_THREADS) return;

    int f   = tid / SEGS;          // feature
    int seg = tid - f * SEGS;      // which float4 chunk of D
    int d0  = seg * 4;

    // Per-feature scalars (25 consecutive threads share f -> broadcast-friendly)
    int2 cc = *reinterpret_cast<const int2*>(coords + 2 * f);   // one b64 load
    int r  = region_ids[f];

    // clip to [WINDOW0, WINDOW1-1]; shift so unsigned / and % match Python floor
    int c0 = min(max(cc.x, -100000), 99999);
    int c1 = min(max(cc.y, -100000), 99999);
    unsigned u[2];
    u[0] = (unsigned)(c0 + 100000);   // [0, 199999]
    u[1] = (unsigned)(c1 + 100000);

    const float* __restrict__ Wr = W + (size_t)r * (size_t)(NW * D_DIM);

    // acc = bias[r, d0:d0+4]  (16B-aligned: r*400 and d0*4 are multiples of 16)
    float4 acc = *reinterpret_cast<const float4*>(bias + (size_t)r * D_DIM + d0);

    const unsigned bw[6]    = {100u, 200u, 500u, 1000u, 2000u, 5000u};
    const int      start[6] = {0, 2001, 3002, 3403, 3604, 3705};
    const float    invbw[6] = {1.0f/100.0f, 1.0f/200.0f, 1.0f/500.0f,
                               1.0f/1000.0f, 1.0f/2000.0f, 1.0f/5000.0f};

    #pragma unroll
    for (int s = 0; s < 6; ++s) {
        #pragma unroll
        for (int c = 0; c < 2; ++c) {
            unsigned q   = u[c] / bw[s];          // constant divisor -> mulhi+shift
            unsigned rem = u[c] - q * bw[s];
            float    a   = (float)rem * invbw[s]; // alpha in [0,1)
            float    om  = 1.0f - a;
            int      bin = (int)q + start[s];     // always in-range (see analysis)

            const float* row = Wr + (size_t)bin * D_DIM + d0;
            // two adjacent weight rows, 400B apart; both 16B-aligned b128 loads;
            // the +400B of the second row folds into the load immediate offset.
            float4 w0 = *reinterpret_cast<const float4*>(row);
            float4 w1 = *reinterpret_cast<const float4*>(row + D_DIM);

            acc.x = fmaf(w0.x, om, fmaf(w1.x, a, acc.x));
            acc.y = fmaf(w0.y, om, fmaf(w1.y, a, acc.y));
            acc.z = fmaf(w0.z, om, fmaf(w1.z, a, acc.z));
            acc.w = fmaf(w0.w, om, fmaf(w1.w, a, acc.w));
        }
    }

    // out[f, d0:d0+4]; f*400 and d0*4 multiples of 16 -> b128 store
    *reinterpret_cast<float4*>(out + (size_t)f * D_DIM + d0) = acc;
}

extern "C" void kernel_launch(void* const* d_in, const int* in_sizes, int n_in,
                              void* d_out, int out_size, void* d_ws, size_t ws_size,
                              hipStream_t stream) {
    (void)in_sizes; (void)n_in; (void)d_ws; (void)ws_size; (void)out_size;

    const int*   coords     = (const int*)  d_in[0];  // coordinates [F,2] int32
    const int*   region_ids = (const int*)  d_in[1];  // region_ids  [F]   int32
    const float* W          = (const float*)d_in[2];  // W [R,NW,D] f32
    const float* bias       = (const float*)d_in[3];  // bias [R,D] f32
    float*       out        = (float*)      d_out;    // [F,D] f32

    const int block = 256;
    const int grid  = (TOTAL_THREADS + block - 1) / block;
    spline_gather_kernel<<<grid, block, 0, stream>>>(coords, region_ids, W, bias, out);
}